// SamplerBase_70463233458398
// MI455X (gfx1250) — compile-verified
//
#include <hip/hip_runtime.h>
#include <hip/hip_bf16.h>
#include <float.h>
#include <limits.h>

// Sampler: top-p (0.9) -> top-k (k from input) -> Gumbel-max categorical.
// Kept set == sorted prefix of length L = min(nucleus_len, k) <= k, so we never sort V.
// Pass 1: async-to-LDS staged stream -> online (max, sumexp) + per-wave bank-skewed
//         1024-bin bit-key histograms (kills same-address ds_add serialization).
// Pass 2: (L2-resident re-read) collect top-k superset candidates, bitonic sort,
//         nucleus walk, gather gumbel at <=k indices, argmax, write outputs.

#define BD      256      // threads per block (8 wave32 waves)
#define NWAVES  (BD / 32)
#define NBUF    4        // async staging buffers (3-deep pipeline)
#define NBINS   1024     // histogram bins (top 10 bits of order-preserving key)
#define HSTRIDE 1032     // per-wave histogram stride: 1032 % 64 == 8 -> bank-skewed copies
#define CAP     1024     // candidate buffer capacity
#define TOPP    0.9f

__device__ __forceinline__ unsigned ordered_key(float x) {
  unsigned b = __float_as_uint(x);
  unsigned m = (unsigned)(((int)b) >> 31) | 0x80000000u;  // neg -> ~b, pos -> b|0x8000..
  return b ^ m;
}

__global__ __launch_bounds__(BD) void sampler_kernel(
    const float* __restrict__ logits,
    const float* __restrict__ gumbel,
    const int*   __restrict__ top_k,
    float* __restrict__ out_conf,
    int*   __restrict__ out_idx,
    float* __restrict__ out_conf2,
    int V)
{
  __shared__ float4   stage[NBUF][BD];          // 16 KB async staging
  __shared__ unsigned histW[NWAVES * HSTRIDE];  // 33 KB per-wave histograms (bank-skewed)
  __shared__ float    cval[CAP];                // 4 KB
  __shared__ int      cidx[CAP];                // 4 KB
  __shared__ float    red_m[NWAVES], red_s[NWAVES];
  __shared__ float    sh_M, sh_S, sh_D;
  __shared__ int      sh_B, sh_L;
  __shared__ unsigned sh_cnt;
  __shared__ float    rv[BD];
  __shared__ int      ri[BD];
  __shared__ float    rc[BD];

  const int row = blockIdx.x;
  const int tid = threadIdx.x;
  const int wid = tid >> 5;
  const int kk  = min(max(*top_k, 1), CAP);
  const size_t rowOff = (size_t)row * (size_t)V;
  const float* rowPtr = logits + rowOff;
  const int V4 = V >> 2;
  const int nStages = (V4 + BD - 1) / BD;

  for (int b = tid; b < NWAVES * HSTRIDE; b += BD) histW[b] = 0u;
  if (tid == 0) sh_cnt = 0u;
  __syncthreads();

  // LDS byte offset of staging buffer (low 32 bits of flat shared address == LDS offset)
  const unsigned ldsBase = (unsigned)(unsigned long long)(&stage[0][0]);

  // ---------------- Pass 1: async-staged stream, online (M,S) + histogram -------------
  for (int s = 0; s < 3 && s < nStages; ++s) {          // prologue: 3 stages in flight
    int f4 = s * BD + tid;
    if (f4 < V4) {
      unsigned loff = ldsBase + (unsigned)(((s & (NBUF - 1)) * BD + tid) * 16);
      unsigned goff = (unsigned)(f4 * 16);
      asm volatile("global_load_async_to_lds_b128 %0, %1, %2"
                   :: "v"(loff), "v"(goff), "s"(rowPtr) : "memory");
    }
  }

  float m = -FLT_MAX, ssum = 0.0f;
  unsigned* myHist = &histW[wid * HSTRIDE];

  for (int i = 0; i < nStages; ++i) {
    const int rem = nStages - 1 - i;
    if (rem >= 2)      asm volatile("s_wait_asynccnt 0x2" ::: "memory");
    else if (rem == 1) asm volatile("s_wait_asynccnt 0x1" ::: "memory");
    else               asm volatile("s_wait_asynccnt 0x0" ::: "memory");

    const int f4 = i * BD + tid;
    if (f4 < V4) {
      float4 v = stage[i & (NBUF - 1)][tid];
      float x4[4] = {v.x, v.y, v.z, v.w};
#pragma unroll
      for (int c = 0; c < 4; ++c) {
        float xv = x4[c];
        if (!(xv >= -FLT_MAX && xv <= FLT_MAX)) xv = -FLT_MAX;  // sanitize non-finite
        atomicAdd(&myHist[ordered_key(xv) >> 22], 1u);          // per-wave ds_add_u32
        if (xv > m) { ssum = ssum * __expf(m - xv) + 1.0f; m = xv; }
        else        { ssum += __expf(xv - m); }
      }
    }
    const int s2 = i + 3;
    if (s2 < nStages) {
      int f42 = s2 * BD + tid;
      if (f42 < V4) {
        unsigned loff = ldsBase + (unsigned)(((s2 & (NBUF - 1)) * BD + tid) * 16);
        unsigned goff = (unsigned)(f42 * 16);
        // dscnt wait: prior ds_loads of the reused buffer must complete before overwrite
        asm volatile("s_wait_dscnt 0x0\n\t"
                     "global_load_async_to_lds_b128 %0, %1, %2"
                     :: "v"(loff), "v"(goff), "s"(rowPtr) : "memory");
      }
    }
  }

  // wave32 reduction of (m, ssum), then cross-wave via LDS
#pragma unroll
  for (int off = 16; off > 0; off >>= 1) {
    float mo = __shfl_down(m, off, 32);
    float so = __shfl_down(ssum, off, 32);
    float mn = fmaxf(m, mo);
    ssum = ssum * __expf(m - mn) + so * __expf(mo - mn);
    m = mn;
  }
  if ((tid & 31) == 0) { red_m[wid] = m; red_s[wid] = ssum; }
  __syncthreads();
  if (tid == 0) {
    float Mv = red_m[0], Sv = red_s[0];
    for (int w = 1; w < NWAVES; ++w) {
      float mo = red_m[w], so = red_s[w];
      float mn = fmaxf(Mv, mo);
      Sv = Sv * __expf(Mv - mn) + so * __expf(mo - mn);
      Mv = mn;
    }
    sh_M = Mv; sh_S = Sv;
  }
  __syncthreads();
  const float M = sh_M, S = sh_S;

  // ---------------- merge per-wave histograms into histW[0..NBINS) ------------------
  {
    unsigned acc[NBINS / BD];
#pragma unroll
    for (int j = 0; j < NBINS / BD; ++j) {
      int b = tid + j * BD;
      unsigned t = 0u;
      for (int w = 0; w < NWAVES; ++w) t += histW[w * HSTRIDE + b];
      acc[j] = t;
    }
    __syncthreads();
#pragma unroll
    for (int j = 0; j < NBINS / BD; ++j) histW[tid + j * BD] = acc[j];
    __syncthreads();
  }

  // ---------------- suffix scan -> candidate bin threshold B ------------------------
  for (int off = 1; off < NBINS; off <<= 1) {
    unsigned v0[NBINS / BD];
#pragma unroll
    for (int j = 0; j < NBINS / BD; ++j) {
      int b = tid + j * BD;
      unsigned x = histW[b];
      if (b + off < NBINS) x += histW[b + off];
      v0[j] = x;
    }
    __syncthreads();
#pragma unroll
    for (int j = 0; j < NBINS / BD; ++j) histW[tid + j * BD] = v0[j];
    __syncthreads();
  }
  if (tid == 0) sh_B = 0;
  __syncthreads();
#pragma unroll
  for (int j = 0; j < NBINS / BD; ++j) {
    int b = tid + j * BD;
    unsigned sb = histW[b];
    unsigned sn = (b + 1 < NBINS) ? histW[b + 1] : 0u;
    if (sb >= (unsigned)kk && sn < (unsigned)kk) sh_B = b;  // unique crossing point
  }
  __syncthreads();
  const int B = sh_B;

  // ---------------- Pass 2 (L2-resident): collect candidates ------------------------
  for (int e = tid; e < CAP; e += BD) { cval[e] = -FLT_MAX; cidx[e] = INT_MAX; }
  __syncthreads();

  const float4* rowPtr4 = (const float4*)rowPtr;
  for (int f4 = tid; f4 < V4; f4 += BD) {
    float4 v = rowPtr4[f4];
    float x4[4] = {v.x, v.y, v.z, v.w};
#pragma unroll
    for (int c = 0; c < 4; ++c) {
      float xv = x4[c];
      if (!(xv >= -FLT_MAX && xv <= FLT_MAX)) xv = -FLT_MAX;
      if ((int)(ordered_key(xv) >> 22) >= B) {
        unsigned pos = atomicAdd(&sh_cnt, 1u);
        if (pos < CAP) { cval[pos] = xv; cidx[pos] = f4 * 4 + c; }
      }
    }
  }
  __syncthreads();
  const int C = (int)min(sh_cnt, (unsigned)CAP);

  // ---------------- bitonic sort (descending, idx tiebreak) over CAP ----------------
  for (int k2 = 2; k2 <= CAP; k2 <<= 1) {
    for (int j = k2 >> 1; j > 0; j >>= 1) {
      for (int e = tid; e < CAP; e += BD) {
        int p = e ^ j;
        if (p > e) {
          float va = cval[e], vb = cval[p];
          int   ia = cidx[e], ib = cidx[p];
          bool aGreater = (va > vb) || (va == vb && ia < ib);
          bool dirDesc  = ((e & k2) == 0);
          if (dirDesc ? (!aGreater) : aGreater) {
            cval[e] = vb; cval[p] = va; cidx[e] = ib; cidx[p] = ia;
          }
        }
      }
      __syncthreads();
    }
  }

  // ---------------- nucleus walk: L = min(nucleus_len, k); D = kept denom -----------
  if (tid == 0) {
    float cum = 0.0f, D = 0.0f;
    int L = 0;
    const int lim = min(C, kk);
    for (int j = 0; j < lim; ++j) {
      if (j > 0 && cum > TOPP) break;      // keep j iff j==0 or cum_{<j} <= TOP_P
      float e = __expf(cval[j] - M);
      D += e; L++;
      cum += e / S;
    }
    sh_D = D; sh_L = L;
  }
  __syncthreads();
  const float Dden = sh_D;
  const int   L    = sh_L;

  // ---------------- gumbel gather (<=k loads) + argmax ------------------------------
  float bv = -FLT_MAX; int bi = INT_MAX; float bl = -FLT_MAX;
  const float* grow = gumbel + rowOff;
  for (int j = tid; j < L; j += BD) {
    int   idx = cidx[j];
    float sc  = cval[j] + grow[idx];
    if (sc > bv || (sc == bv && idx < bi)) { bv = sc; bi = idx; bl = cval[j]; }
  }
  rv[tid] = bv; ri[tid] = bi; rc[tid] = bl;
  __syncthreads();
  if (tid == 0) {
    for (int t = 1; t < BD; ++t) {
      if (rv[t] > bv || (rv[t] == bv && ri[t] < bi)) { bv = rv[t]; bi = ri[t]; bl = rc[t]; }
    }
    float conf = __expf(bl - M) / Dden;
    out_conf[row]  = conf;   // confidence
    out_idx[row]   = bi;     // x0 (int32 bits)
    out_conf2[row] = conf;   // initial_confidence
  }
}

extern "C" void kernel_launch(void* const* d_in, const int* in_sizes, int n_in,
                              void* d_out, int out_size, void* d_ws, size_t ws_size,
                              hipStream_t stream) {
  (void)n_in; (void)d_ws; (void)ws_size; (void)out_size;
  const float* logits = (const float*)d_in[0];
  const float* gumbel = (const float*)d_in[1];
  const int*   top_k  = (const int*)d_in[2];

  const int V = 128000;
  const int N = in_sizes[0] / V;

  float* out       = (float*)d_out;
  float* out_conf  = out;                 // [N] f32
  int*   out_idx   = (int*)(out + N);     // [N] i32 (bit-stored)
  float* out_conf2 = out + 2 * N;         // [N] f32

  sampler_kernel<<<N, BD, 0, stream>>>(logits, gumbel, top_k,
                                       out_conf, out_idx, out_conf2, V);
}